// EglGCNConvTest_40767829573731
// MI455X (gfx1250) — compile-verified
//
#include <hip/hip_runtime.h>

// ---------------------------------------------------------------------------
// GCN conv: out = (segment_sum over edges of (h@W * norm)[src] into dst) * norm
// N=100000 nodes, 64->64 feats, E=1.6M edges.
//
// Phase traffic: GEMM ~51MB, scatter ~410MB gather + 102M f32 atomics (both
// L2-resident: hn and out are 25.6MB each vs 192MB L2). Scatter dominates;
// GEMM must use v_wmma_f32_16x16x4_f32 to not be compute-limited in its window.
// ---------------------------------------------------------------------------

typedef __attribute__((ext_vector_type(2))) float v2f;
typedef __attribute__((ext_vector_type(8))) float v8f;

// ---- Kernel 1: zero the accumulator (d_out), float4-wide ----
__global__ __launch_bounds__(256) void zero_kernel(float4* __restrict__ out, int n4) {
  int i = blockIdx.x * blockDim.x + threadIdx.x;
  if (i < n4) out[i] = make_float4(0.f, 0.f, 0.f, 0.f);
}

// ---- Kernel 2: hn = (h @ W) * norm, via WMMA f32 16x16x4 ----
// Block = 256 threads = 8 waves; block covers 32 rows x 64 cols.
// Wave w: rowTile = w>>2 (16 rows), colTile = w&3 (16 cols). K=64 in 16 steps.
__global__ __launch_bounds__(256) void gemm_scale_kernel(
    const float* __restrict__ h, const float* __restrict__ w,
    const float* __restrict__ norm, float* __restrict__ hn, int n_nodes) {
  __shared__ float wlds[64 * 64];  // 16KB: whole weight matrix, row-major

  // cooperative stage of W into LDS (float4-wide, 4 per thread)
  {
    const float4* wsrc = (const float4*)w;
    float4* wdst = (float4*)wlds;
    #pragma unroll
    for (int i = threadIdx.x; i < 64 * 64 / 4; i += 256) wdst[i] = wsrc[i];
  }
  __syncthreads();

  const int wave  = threadIdx.x >> 5;
  const int lane  = threadIdx.x & 31;
  const int c0    = (wave & 3) * 16;                    // col tile base
  const int rw    = blockIdx.x * 32 + (wave >> 2) * 16; // row tile base
  const int m     = lane & 15;                          // A: row M, B: col N
  const int khalf = lane >> 4;                          // K sub-pair select

  // A-fragment source row (clamped for safety; N % 32 == 0 so never triggers)
  int arow = rw + m;
  if (arow >= n_nodes) arow = n_nodes - 1;
  const float* aptr = h + arow * 64 + khalf * 2;

  v8f acc = {};
  #pragma unroll
  for (int kk = 0; kk < 16; ++kk) {
    const int kbase = kk * 4 + khalf * 2;
    // A 16x4 layout: lanes 0-15 hold K={kbase,kbase+1} of row m (khalf=0),
    // lanes 16-31 hold K={kbase,kbase+1} with khalf=1 offset. Contiguous pair.
    const float2 av = *(const float2*)(aptr + kk * 4);
    v2f a; a.x = av.x; a.y = av.y;
    // B 4x16 layout (mirror of A): lane holds W[kbase][c0+m], W[kbase+1][c0+m]
    v2f b;
    b.x = wlds[kbase * 64 + c0 + m];
    b.y = wlds[(kbase + 1) * 64 + c0 + m];
    acc = __builtin_amdgcn_wmma_f32_16x16x4_f32(
        /*neg_a=*/false, a, /*neg_b=*/false, b,
        /*c_mod=*/(short)0, acc, /*reuse_a=*/false, /*reuse_b=*/false);
  }

  // D layout: VGPR r -> row (rw + khalf*8 + r), col (c0 + m). Fuse norm scale.
  const int rbase = rw + khalf * 8;
  #pragma unroll
  for (int r = 0; r < 8; ++r) {
    const int row = rbase + r;
    if (row < n_nodes) hn[row * 64 + c0 + m] = acc[r] * norm[row];
  }
}

// ---- Kernel 3: per-edge gather + atomic scatter-add ----
// One wave per edge: src/dst loads coalesce to one request per wave,
// gather is a contiguous 256B float2 read, scatter is 2 no-return
// agent-scope relaxed f32 atomics per lane (resolved in L2).
__global__ __launch_bounds__(256) void scatter_kernel(
    const float* __restrict__ hn, const int* __restrict__ src,
    const int* __restrict__ dst, float* __restrict__ out, int n_edges) {
  const int e = blockIdx.x * 8 + (threadIdx.x >> 5);
  if (e >= n_edges) return;
  const int lane = threadIdx.x & 31;
  const int s = src[e];
  const int d = dst[e];
  const float2 v = *(const float2*)(hn + s * 64 + lane * 2);
  float* p = out + d * 64 + lane * 2;
  (void)__hip_atomic_fetch_add(p,     v.x, __ATOMIC_RELAXED, __HIP_MEMORY_SCOPE_AGENT);
  (void)__hip_atomic_fetch_add(p + 1, v.y, __ATOMIC_RELAXED, __HIP_MEMORY_SCOPE_AGENT);
}

// ---- Kernel 4: in-place post-scale by destination norm, float4-wide ----
__global__ __launch_bounds__(256) void scale_kernel(
    float* __restrict__ out, const float* __restrict__ norm, int n_nodes) {
  const int i = blockIdx.x * blockDim.x + threadIdx.x;  // float4 index
  const int n4 = n_nodes * 16;                          // 64 floats = 16 float4 per row
  if (i >= n4) return;
  const float s = norm[i >> 4];
  float4* p = (float4*)out + i;
  float4 v = *p;
  v.x *= s; v.y *= s; v.z *= s; v.w *= s;
  *p = v;
}

extern "C" void kernel_launch(void* const* d_in, const int* in_sizes, int n_in,
                              void* d_out, int out_size, void* d_ws, size_t ws_size,
                              hipStream_t stream) {
  const float* h    = (const float*)d_in[0];   // [N,64]
  const float* w    = (const float*)d_in[1];   // [64,64]
  const float* norm = (const float*)d_in[2];   // [N,1]
  const int*   src  = (const int*)d_in[3];     // [E]
  const int*   dst  = (const int*)d_in[4];     // [E]
  float* out = (float*)d_out;                  // [N,64]
  float* hn  = (float*)d_ws;                   // scratch: N*64 floats = 25.6MB

  const int n_nodes = in_sizes[2];
  const int n_edges = in_sizes[3];
  const int n4 = n_nodes * 16;

  zero_kernel<<<(n4 + 255) / 256, 256, 0, stream>>>((float4*)out, n4);

  gemm_scale_kernel<<<(n_nodes + 31) / 32, 256, 0, stream>>>(h, w, norm, hn, n_nodes);

  scatter_kernel<<<(n_edges + 7) / 8, 256, 0, stream>>>(hn, src, dst, out, n_edges);

  scale_kernel<<<(n4 + 255) / 256, 256, 0, stream>>>(out, norm, n_nodes);
}